// CRF_40398462386204
// MI455X (gfx1250) — compile-verified
//
#include <hip/hip_runtime.h>
#include <hip/hip_bf16.h>

typedef __attribute__((ext_vector_type(2))) float v2f;
typedef __attribute__((ext_vector_type(8))) float v8f;

#define CRF_T 512
#define CRF_B 64
#define CRF_D 1024
#define CRF_K 10
#define NPAD 16

// ---------------------------------------------------------------------------
// Kernel 1: emissions = word_features @ W + b, padded to 16 tag columns.
// One wave computes one 16x16 output tile via V_WMMA_F32_16X16X4_F32,
// looping k over D=1024 in steps of 4. W (padded to 1024x16) staged in LDS.
// feat is streamed once (134 MB) -> non-temporal loads.
// ---------------------------------------------------------------------------
__global__ __launch_bounds__(256)
void crf_emissions_wmma(const float* __restrict__ feat,   // (T*B, D)
                        const float* __restrict__ W,      // (D, K)
                        const float* __restrict__ bias,   // (K)
                        float* __restrict__ emis)         // (T*B, 16) padded
{
    __shared__ float wlds[CRF_D * NPAD];   // 64 KB: padded W, row-major (k, n)

    for (int i = threadIdx.x; i < CRF_D * NPAD; i += blockDim.x) {
        int k = i >> 4, n = i & 15;
        wlds[i] = (n < CRF_K) ? W[k * CRF_K + n] : 0.0f;
    }
    __syncthreads();

    const int wave = threadIdx.x >> 5;
    const int lane = threadIdx.x & 31;
    const int tile = blockIdx.x * (blockDim.x >> 5) + wave;  // 2048 tiles
    const int rowBase = tile * 16;
    const int mrow = lane & 15;
    const int half = lane >> 4;
    const int n = lane & 15;

    const float* arow = feat + (size_t)(rowBase + mrow) * CRF_D + 2 * half;

    v8f acc = {};
    for (int k = 0; k < CRF_D; k += 4) {
        // A fragment: lane(row, half) holds K = k+2*half, k+2*half+1
        v2f a = __builtin_nontemporal_load((const v2f*)(arow + k));
        // B fragment: lanes 0-15 -> K rows k, k+1 ; lanes 16-31 -> k+2, k+3
        v2f bb;
        bb.x = wlds[(k + 2 * half) * NPAD + n];
        bb.y = wlds[(k + 2 * half + 1) * NPAD + n];
        acc = __builtin_amdgcn_wmma_f32_16x16x4_f32(
            /*neg_a=*/false, a, /*neg_b=*/false, bb,
            /*c_mod=*/(short)0, acc, /*reuse_a=*/false, /*reuse_b=*/false);
    }

    const float bv = (n < CRF_K) ? bias[n] : 0.0f;
    #pragma unroll
    for (int v = 0; v < 8; ++v) {
        int m = v + 8 * half;  // C/D layout: VGPR v holds M=v (lanes<16), M=v+8
        emis[(size_t)(rowBase + m) * NPAD + n] = acc[v] + bv;
    }
}

// ---------------------------------------------------------------------------
// Kernel 2: per-batch CRF, fused scans. One single-wave block per batch elem.
//   lanes  0..9  : forward (logsumexp) state  -> partition function
//   lanes 16..25 : Viterbi (max-plus) state   -> decode history
// The max/argmax loop is shared; only the exp/log tail is LSE-lane-predicated.
// Gold-path numerator is a 32-lane strided reduction.
// ---------------------------------------------------------------------------
__global__ __launch_bounds__(32)
void crf_scan_fused(const float* __restrict__ emis,    // (T*B, 16) padded
                    const int*   __restrict__ words,   // (T, B)
                    const int*   __restrict__ tags,    // (T, B)
                    const float* __restrict__ startv,  // (K)
                    const float* __restrict__ endv,    // (K)
                    const float* __restrict__ trans,   // (K, K)
                    float* __restrict__ paths,         // (T, B) as float
                    float* __restrict__ partial)       // (B): numer - denom
{
    const int b    = blockIdx.x;
    const int lane = threadIdx.x;
    const bool isF = (lane < CRF_K);                        // logsumexp lanes
    const bool isV = (lane >= 16) && (lane < 16 + CRF_K);   // Viterbi lanes
    const int  j   = isV ? (lane - 16) : (isF ? lane : 0);  // state index

    __shared__ float sF[16];
    __shared__ float sV[16];
    __shared__ float tr[CRF_K * CRF_K];
    __shared__ unsigned char msk[CRF_T];
    __shared__ int   hist[(CRF_T - 1) * CRF_K];
    __shared__ float red[32];
    __shared__ int   redi[32];

    for (int i = lane; i < CRF_K * CRF_K; i += 32) tr[i] = trans[i];
    for (int t = lane; t < CRF_T; t += 32) msk[t] = (words[t * CRF_B + b] != 0);

    // per-lane register copy of transition column j
    float trc[CRF_K];
    #pragma unroll
    for (int i = 0; i < CRF_K; ++i) trc[i] = trans[i * CRF_K + j];

    // init both states
    if (isF) sF[j] = startv[j] + emis[(size_t)b * NPAD + j];
    if (isV) sV[j] = startv[j] + emis[(size_t)b * NPAD + j];
    __syncthreads();

    // ---- fused forward + Viterbi scan ----
    float em_cur = emis[((size_t)1 * CRF_B + b) * NPAD + j];
    for (int t = 1; t < CRF_T; ++t) {
        // prefetch next step's emission early (latency hidden by step compute)
        float em_nxt = (t + 1 < CRF_T)
                         ? emis[((size_t)(t + 1) * CRF_B + b) * NPAD + j] : 0.0f;

        const float* src = isV ? sV : sF;
        float sv[CRF_K];
        #pragma unroll
        for (int i = 0; i < CRF_K; ++i) sv[i] = src[i];

        // shared max/argmax (LSE needs max for stability; Viterbi needs both)
        float m = -1e30f; int bi = 0;
        #pragma unroll
        for (int i = 0; i < CRF_K; ++i) {
            float v = sv[i] + trc[i];
            if (v > m) { m = v; bi = i; }
        }
        float nv;
        if (isF) {
            float sum = 0.0f;
            #pragma unroll
            for (int i = 0; i < CRF_K; ++i) sum += __expf(sv[i] + trc[i] - m);
            nv = m + __logf(sum) + em_cur;
        } else {
            nv = m + em_cur;
        }
        if (isV) hist[(t - 1) * CRF_K + j] = bi;  // stored regardless of mask

        __syncthreads();
        if (msk[t]) {
            if (isF) sF[j] = nv;
            else if (isV) sV[j] = nv;
        }
        __syncthreads();
        em_cur = em_nxt;
    }

    // ---- gold-path numerator: strided 32-lane reduction ----
    float nsum = 0.0f; int smax = 0;
    for (int t = 1 + lane; t < CRF_T; t += 32) {
        if (msk[t]) {
            int pt = tags[(t - 1) * CRF_B + b];
            int ct = tags[t * CRF_B + b];
            nsum += tr[pt * CRF_K + ct] +
                    emis[((size_t)t * CRF_B + b) * NPAD + ct];
            smax = t;  // strided ascending -> last masked t seen by this lane
        }
    }
    red[lane] = nsum;
    redi[lane] = smax;
    __syncthreads();

    if (lane == 0) {
        float numer = 0.0f; int seqEnd = 0;
        for (int i = 0; i < 32; ++i) {
            numer += red[i];
            seqEnd = (redi[i] > seqEnd) ? redi[i] : seqEnd;
        }
        int t0 = tags[b];
        numer += startv[t0] + emis[(size_t)b * NPAD + t0];
        numer += endv[tags[seqEnd * CRF_B + b]];

        // denominator from forward state
        float m = -1e30f;
        for (int i = 0; i < CRF_K; ++i) m = fmaxf(m, sF[i] + endv[i]);
        float sum = 0.0f;
        for (int i = 0; i < CRF_K; ++i) sum += __expf(sF[i] + endv[i] - m);
        partial[b] = numer - (m + __logf(sum));

        // Viterbi terminal tag + backtrack over LDS history
        float bm = -1e30f; int tag = 0;
        for (int i = 0; i < CRF_K; ++i) {
            float v = sV[i] + endv[i];
            if (v > bm) { bm = v; tag = i; }
        }
        paths[(size_t)(CRF_T - 1) * CRF_B + b] =
            msk[CRF_T - 1] ? (float)tag : 0.0f;
        for (int t = CRF_T - 2; t >= 0; --t) {
            int prev = msk[t + 1] ? hist[t * CRF_K + tag] : tag;
            tag = prev;
            paths[(size_t)t * CRF_B + b] = msk[t] ? (float)prev : 0.0f;
        }
    }
}

// ---------------------------------------------------------------------------
// Kernel 3: loss = -sum_b (numer_b - denom_b)
// ---------------------------------------------------------------------------
__global__ void crf_loss_reduce(const float* __restrict__ partial,
                                float* __restrict__ out)
{
    if (threadIdx.x == 0 && blockIdx.x == 0) {
        float sum = 0.0f;
        for (int i = 0; i < CRF_B; ++i) sum += partial[i];
        out[CRF_T * CRF_B] = -sum;
    }
}

extern "C" void kernel_launch(void* const* d_in, const int* in_sizes, int n_in,
                              void* d_out, int out_size, void* d_ws, size_t ws_size,
                              hipStream_t stream) {
    const int*   words  = (const int*)d_in[0];
    const int*   tags   = (const int*)d_in[1];
    const float* feat   = (const float*)d_in[2];
    const float* W      = (const float*)d_in[3];
    const float* bias   = (const float*)d_in[4];
    const float* startv = (const float*)d_in[5];
    const float* endv   = (const float*)d_in[6];
    const float* trans  = (const float*)d_in[7];

    float* out = (float*)d_out;

    // workspace: padded emissions (T*B*16 floats = 2 MB) + partials (B floats)
    float* emis    = (float*)d_ws;
    float* partial = emis + (size_t)CRF_T * CRF_B * NPAD;

    // 2048 tiles of 16 rows, 8 waves (256 threads) per block -> 256 blocks
    crf_emissions_wmma<<<256, 256, 0, stream>>>(feat, W, bias, emis);
    crf_scan_fused<<<CRF_B, 32, 0, stream>>>(emis, words, tags, startv, endv,
                                             trans, out, partial);
    crf_loss_reduce<<<1, 32, 0, stream>>>(partial, out);
}